// structure_attention_net_57260503990855
// MI455X (gfx1250) — compile-verified
//
#include <hip/hip_runtime.h>
#include <math.h>

// ---------------------------------------------------------------------------
// StructureAttentionNet forward on MI455X (gfx1250), fp32 WMMA path.
// Shapes: N=17 nodes (padded to 32), D=512, H=8 heads, hidden=256, cat=515.
// All GEMMs use V_WMMA_F32_16X16X4_F32 (wave32, 16x16 f32 tiles).
// Guarded loads are expressed as clamp-index + mask-multiply so no EXEC
// divergence appears inside the WMMA K-loops.
// ---------------------------------------------------------------------------

#define NN      17
#define MPAD    32
#define DD      512
#define HH      8
#define HID     256
#define DC      515   // D + 3 coords

typedef __attribute__((ext_vector_type(2))) float v2f;
typedef __attribute__((ext_vector_type(8))) float v8f;

__device__ __forceinline__ v8f wmma4(v2f a, v2f b, v8f c) {
  // 8-arg form: (neg_a, A, neg_b, B, c_mod, C, reuse_a, reuse_b)
  return __builtin_amdgcn_wmma_f32_16x16x4_f32(false, a, false, b, (short)0, c,
                                               false, false);
}

__device__ __forceinline__ v2f ld2(const float* p) {  // 8B-aligned pair load
  return *(const v2f*)p;
}

// ---------------------------------------------------------------------------
// K1: f[h][m][d] = relu( Xs[m][:] @ FC[h][:][d] ), m padded to 32 (zeros).
// Grid (4, 8): x = 128-col slab, y = head. Block = 128 thr = 4 waves,
// each wave computes a 32x32 output tile (2 Mtiles x 2 Ntiles).
// ---------------------------------------------------------------------------
__global__ void k_gemm_f(const float* __restrict__ Xs,
                         const float* __restrict__ FC,
                         float* __restrict__ f_ws) {
  const int h    = blockIdx.y;
  const int wave = threadIdx.x >> 5;
  const int lane = threadIdx.x & 31;
  const int half = lane >> 4;
  const int l16  = lane & 15;
  const int d0   = blockIdx.x * 128 + wave * 32;
  const float* B = FC + (size_t)h * DD * DD;

  // Tile0 rows 0..15 all valid; tile1 rows 16..31 clamp to row 16, masked.
  const float* a0p = Xs + (size_t)l16 * DD;
  const float* a1p = Xs + (size_t)16 * DD;         // clamp(16+l16, 16)
  const float  mk1 = (l16 == 0) ? 1.0f : 0.0f;

  v8f acc[2][2] = {};
  for (int k = 0; k < DD; k += 4) {
    const int kk = k + 2 * half;
    v2f a0 = ld2(a0p + kk);
    v2f a1 = ld2(a1p + kk);
    a1.x *= mk1; a1.y *= mk1;
#pragma unroll
    for (int u = 0; u < 2; ++u) {
      const int n = d0 + u * 16 + l16;
      v2f b;
      b.x = B[(size_t)kk * DD + n];
      b.y = B[(size_t)(kk + 1) * DD + n];
      acc[0][u] = wmma4(a0, b, acc[0][u]);
      acc[1][u] = wmma4(a1, b, acc[1][u]);
    }
  }
#pragma unroll
  for (int t = 0; t < 2; ++t)
#pragma unroll
    for (int u = 0; u < 2; ++u)
#pragma unroll
      for (int r = 0; r < 8; ++r) {
        const int m = t * 16 + r + 8 * half;
        const int n = d0 + u * 16 + l16;
        const float v = acc[t][u][r];
        f_ws[((size_t)h * MPAD + m) * DD + n] = v > 0.0f ? v : 0.0f;
      }
}

// ---------------------------------------------------------------------------
// K2: hdn[h][m][n] = tanh( [f|ROIs][m][:] @ W1[h][:][n] + b1[h][n] )
// K = 512 (from f_ws, padded rows already zero) + one padded chunk of 3 coords.
// Grid (2, 8), block 128.
// ---------------------------------------------------------------------------
__global__ void k_gemm_hdn(const float* __restrict__ f_ws,
                           const float* __restrict__ ROIs,
                           const float* __restrict__ W1,
                           const float* __restrict__ b1,
                           float* __restrict__ hdn_ws) {
  const int h    = blockIdx.y;
  const int wave = threadIdx.x >> 5;
  const int lane = threadIdx.x & 31;
  const int half = lane >> 4;
  const int l16  = lane & 15;
  const int n0   = blockIdx.x * 128 + wave * 32;
  const float* A = f_ws + (size_t)h * MPAD * DD;
  const float* B = W1 + (size_t)h * DC * HID;
  const float* a0p = A + (size_t)l16 * DD;
  const float* a1p = A + (size_t)(16 + l16) * DD;  // padded rows are zeros

  v8f acc[2][2] = {};
  for (int k = 0; k < DD; k += 4) {
    const int kk = k + 2 * half;
    v2f a0 = ld2(a0p + kk);
    v2f a1 = ld2(a1p + kk);
#pragma unroll
    for (int u = 0; u < 2; ++u) {
      const int n = n0 + u * 16 + l16;
      v2f b;
      b.x = B[(size_t)kk * HID + n];
      b.y = B[(size_t)(kk + 1) * HID + n];
      acc[0][u] = wmma4(a0, b, acc[0][u]);
      acc[1][u] = wmma4(a1, b, acc[1][u]);
    }
  }
  // coordinate chunk: K columns 512..514 real, column 515 zero-padded via A.
  {
    const int kk = DD + 2 * half;                 // 512 or 514
    const int c0 = kk - DD;                       // 0 or 2 (always valid)
    const int c1v = (c0 + 1 < 3) ? (c0 + 1) : 0;  // clamped coord index
    const float mc1 = (c0 + 1 < 3) ? 1.0f : 0.0f; // zero for padded col 515
    // tile0 rows 0..15 valid; tile1: clamp to row 16, mask by (l16==0)
    const float mk1 = (l16 == 0) ? 1.0f : 0.0f;
    v2f a0, a1;
    a0.x = ROIs[l16 * 3 + c0];
    a0.y = ROIs[l16 * 3 + c1v] * mc1;
    a1.x = ROIs[16 * 3 + c0] * mk1;
    a1.y = ROIs[16 * 3 + c1v] * (mc1 * mk1);
    // B row for padded column clamped to a valid row (A supplies the zero).
    const int r1 = (kk + 1 < DC) ? (kk + 1) : kk;
#pragma unroll
    for (int u = 0; u < 2; ++u) {
      const int n = n0 + u * 16 + l16;
      v2f b;
      b.x = B[(size_t)kk * HID + n];
      b.y = B[(size_t)r1 * HID + n];
      acc[0][u] = wmma4(a0, b, acc[0][u]);
      acc[1][u] = wmma4(a1, b, acc[1][u]);
    }
  }
#pragma unroll
  for (int t = 0; t < 2; ++t)
#pragma unroll
    for (int u = 0; u < 2; ++u)
#pragma unroll
      for (int r = 0; r < 8; ++r) {
        const int m = t * 16 + r + 8 * half;
        const int n = n0 + u * 16 + l16;
        hdn_ws[((size_t)h * MPAD + m) * HID + n] =
            tanhf(acc[t][u][r] + b1[h * HID + n]);
      }
}

// ---------------------------------------------------------------------------
// K3: att[h][m][k] = hdn[h][m][:] @ W2[h][:][k] + b2[h][k]; k padded to 32.
// Grid 8 blocks x 1 wave; each wave does the whole 32x32 tile.
// ---------------------------------------------------------------------------
__global__ void k_gemm_att(const float* __restrict__ hdn_ws,
                           const float* __restrict__ W2,
                           const float* __restrict__ b2,
                           float* __restrict__ att_ws) {
  const int h    = blockIdx.x;
  const int lane = threadIdx.x & 31;
  const int half = lane >> 4;
  const int l16  = lane & 15;
  const float* A = hdn_ws + (size_t)h * MPAD * HID;
  const float* B = W2 + (size_t)h * HID * NN;
  const float* a0p = A + (size_t)l16 * HID;
  const float* a1p = A + (size_t)(16 + l16) * HID;

  // Column guards (loop-invariant): clamp n to 16 and mask.
  int  nc[2];  float nm[2];
#pragma unroll
  for (int u = 0; u < 2; ++u) {
    const int n = u * 16 + l16;
    nc[u] = (n < NN) ? n : (NN - 1);
    nm[u] = (n < NN) ? 1.0f : 0.0f;
  }

  v8f acc[2][2] = {};
  for (int k = 0; k < HID; k += 4) {
    const int kk = k + 2 * half;
    v2f a0 = ld2(a0p + kk);
    v2f a1 = ld2(a1p + kk);
#pragma unroll
    for (int u = 0; u < 2; ++u) {
      v2f b;
      b.x = B[kk * NN + nc[u]] * nm[u];
      b.y = B[(kk + 1) * NN + nc[u]] * nm[u];
      acc[0][u] = wmma4(a0, b, acc[0][u]);
      acc[1][u] = wmma4(a1, b, acc[1][u]);
    }
  }
#pragma unroll
  for (int t = 0; t < 2; ++t)
#pragma unroll
    for (int u = 0; u < 2; ++u) {
      const float bias = b2[h * NN + nc[u]] * nm[u];
#pragma unroll
      for (int r = 0; r < 8; ++r) {
        const int m = t * 16 + r + 8 * half;
        const int n = u * 16 + l16;
        att_ws[((size_t)h * MPAD + m) * MPAD + n] = acc[t][u][r] + bias;
      }
    }
}

// ---------------------------------------------------------------------------
// K4: softmax over the node axis. atts[h][k][n] = softmax_n att[h][n][k].
// Padded landmarks/nodes written as exact zeros (so K5 needs no masking).
// One thread per (h, k): 256 threads, 1 block.
// ---------------------------------------------------------------------------
__global__ void k_softmax(const float* __restrict__ att_ws,
                          float* __restrict__ atts_ws) {
  const int idx = threadIdx.x;          // 0..255
  const int h   = idx >> 5;
  const int kl  = idx & 31;
  float* dst = atts_ws + ((size_t)h * MPAD + kl) * MPAD;
  if (kl < NN) {
    float v[NN];
    float mx = -3.402823466e+38f;
    for (int n = 0; n < NN; ++n) {
      v[n] = att_ws[((size_t)h * MPAD + n) * MPAD + kl];
      mx = v[n] > mx ? v[n] : mx;
    }
    float s = 0.0f;
    for (int n = 0; n < NN; ++n) { v[n] = __expf(v[n] - mx); s += v[n]; }
    const float inv = 1.0f / s;
    for (int n = 0; n < NN; ++n) dst[n] = v[n] * inv;
    for (int n = NN; n < MPAD; ++n) dst[n] = 0.0f;
  } else {
    for (int n = 0; n < MPAD; ++n) dst[n] = 0.0f;
  }
}

// ---------------------------------------------------------------------------
// K5: h_prime[h][k][d] = atts[h][k][:] @ f[h][:][d].  M=32, K=32, N=512.
// Grid (4, 8), block 128.
// ---------------------------------------------------------------------------
__global__ void k_gemm_hp(const float* __restrict__ atts_ws,
                          const float* __restrict__ f_ws,
                          float* __restrict__ hp_ws) {
  const int h    = blockIdx.y;
  const int wave = threadIdx.x >> 5;
  const int lane = threadIdx.x & 31;
  const int half = lane >> 4;
  const int l16  = lane & 15;
  const int d0   = blockIdx.x * 128 + wave * 32;
  const float* A = atts_ws + (size_t)h * MPAD * MPAD;
  const float* B = f_ws + (size_t)h * MPAD * DD;
  const float* a0p = A + (size_t)l16 * MPAD;       // padded rows are zeros
  const float* a1p = A + (size_t)(16 + l16) * MPAD;

  v8f acc[2][2] = {};
#pragma unroll
  for (int k = 0; k < MPAD; k += 4) {
    const int kk = k + 2 * half;
    v2f a0 = ld2(a0p + kk);
    v2f a1 = ld2(a1p + kk);
#pragma unroll
    for (int u = 0; u < 2; ++u) {
      const int n = d0 + u * 16 + l16;
      v2f b;
      b.x = B[(size_t)kk * DD + n];
      b.y = B[(size_t)(kk + 1) * DD + n];
      acc[0][u] = wmma4(a0, b, acc[0][u]);
      acc[1][u] = wmma4(a1, b, acc[1][u]);
    }
  }
#pragma unroll
  for (int t = 0; t < 2; ++t)
#pragma unroll
    for (int u = 0; u < 2; ++u)
#pragma unroll
      for (int r = 0; r < 8; ++r) {
        const int m = t * 16 + r + 8 * half;
        const int n = d0 + u * 16 + l16;
        hp_ws[((size_t)h * MPAD + m) * DD + n] = acc[t][u][r];
      }
}

// ---------------------------------------------------------------------------
// K6: out = relu(sum_h conv_w[h]*h_prime[h] + conv_b) + Xs; plus ROI copy.
// ---------------------------------------------------------------------------
__global__ void k_combine(const float* __restrict__ hp_ws,
                          const float* __restrict__ Xs,
                          const float* __restrict__ ROIs,
                          const float* __restrict__ conv_w,
                          const float* __restrict__ conv_b,
                          float* __restrict__ out) {
  const int idx = blockIdx.x * 256 + threadIdx.x;
  if (idx < NN * DD) {
    const int k = idx / DD;
    const int d = idx % DD;
    float s = conv_b[0];
#pragma unroll
    for (int h = 0; h < HH; ++h)
      s += conv_w[h] * hp_ws[((size_t)h * MPAD + k) * DD + d];
    const float v = s > 0.0f ? s : 0.0f;
    out[idx] = v + Xs[idx];
  } else if (idx < NN * DD + NN * 3) {
    out[idx] = ROIs[idx - NN * DD];
  }
}

// ---------------------------------------------------------------------------
extern "C" void kernel_launch(void* const* d_in, const int* in_sizes, int n_in,
                              void* d_out, int out_size, void* d_ws,
                              size_t ws_size, hipStream_t stream) {
  (void)in_sizes; (void)n_in; (void)out_size; (void)ws_size;
  const float* Xs   = (const float*)d_in[0];   // [17,512]
  const float* ROIs = (const float*)d_in[1];   // [17,3]
  // d_in[2] adj unused
  const float* FC   = (const float*)d_in[3];   // [8,512,512]
  const float* W1   = (const float*)d_in[4];   // [8,515,256]
  const float* b1   = (const float*)d_in[5];   // [8,256]
  const float* W2   = (const float*)d_in[6];   // [8,256,17]
  const float* b2   = (const float*)d_in[7];   // [8,17]
  const float* cw   = (const float*)d_in[8];   // [8]
  const float* cb   = (const float*)d_in[9];   // [1]
  float* out = (float*)d_out;

  float* ws      = (float*)d_ws;
  float* f_ws    = ws;                         // 8*32*512 = 131072
  float* hdn_ws  = ws + 131072;                // 8*32*256 =  65536
  float* att_ws  = ws + 131072 + 65536;        // 8*32*32  =   8192
  float* atts_ws = ws + 131072 + 65536 + 8192; // 8*32*32  =   8192
  float* hp_ws   = ws + 131072 + 65536 + 8192 + 8192; // 131072

  k_gemm_f  <<<dim3(4, 8), 128, 0, stream>>>(Xs, FC, f_ws);
  k_gemm_hdn<<<dim3(2, 8), 128, 0, stream>>>(f_ws, ROIs, W1, b1, hdn_ws);
  k_gemm_att<<<8, 32, 0, stream>>>(hdn_ws, W2, b2, att_ws);
  k_softmax <<<1, 256, 0, stream>>>(att_ws, atts_ws);
  k_gemm_hp <<<dim3(4, 8), 128, 0, stream>>>(atts_ws, f_ws, hp_ws);
  k_combine <<<35, 256, 0, stream>>>(hp_ws, Xs, ROIs, cw, cb, out);
}